// GNNEncoder_32323923870319
// MI455X (gfx1250) — compile-verified
//
#include <hip/hip_runtime.h>
#include <hip/hip_bf16.h>

typedef __attribute__((ext_vector_type(2))) float v2f;
typedef __attribute__((ext_vector_type(8))) float v8f;

#define D 64
#define WAVES_PER_BLOCK 8

// ---------------------------------------------------------------------------
// Zero out msg_sum + deg scratch (N*(D+1) floats).
// ---------------------------------------------------------------------------
__global__ void zero_f32(float* __restrict__ p, long long n) {
  long long i = (long long)blockIdx.x * blockDim.x + threadIdx.x;
  if (i < n) p[i] = 0.0f;
}

// ---------------------------------------------------------------------------
// Edge scatter: 64 threads per edge. Coalesced 256B gather of feat[src],
// float atomics into msg_sum[dst] (L2-resident: both arrays ~25.6MB << 192MB).
// Lane f==0 also bumps the degree counter.
// ---------------------------------------------------------------------------
__global__ void scatter_mean_accum(const float* __restrict__ feat,
                                   const long long* __restrict__ srcIdx,
                                   const long long* __restrict__ dstIdx,
                                   float* __restrict__ msum,
                                   float* __restrict__ deg,
                                   long long nEdges) {
  long long idx = (long long)blockIdx.x * blockDim.x + threadIdx.x;
  long long e = idx >> 6;           // D == 64 features per edge
  int f = (int)(idx & 63);
  if (e >= nEdges) return;
  long long s = srcIdx[e];
  long long d = dstIdx[e];
  atomicAdd(&msum[d * D + f], feat[s * D + f]);
  if (f == 0) atomicAdd(&deg[d], 1.0f);
}

// ---------------------------------------------------------------------------
// Fused SAGE layer GEMM in exact fp32 via V_WMMA_F32_16X16X4_F32:
//   out = (msum/deg) @ wl + bias + xin @ wr   [+ ReLU]
// One wave per 16-row tile; weights staged in LDS.
//
// A-fragment (16x4 f32): lane 0-15 holds row M=lane, {K=kb, K=kb+1};
//                        lane 16-31 holds row M=lane-16, {K=kb+2, K=kb+3}.
// B-fragment (4x16 f32): lane 0-15 holds col N=lane, {K=kb, kb+1};
//                        lane 16-31 holds col N=lane-16, {K=kb+2, kb+3}.
// C/D (16x16 f32): VGPR r, lanes 0-15 -> (M=r, N=lane);
//                  lanes 16-31 -> (M=8+r, N=lane-16).
// ---------------------------------------------------------------------------
template <bool RELU>
__global__ __launch_bounds__(256) void sage_wmma_gemm(
    const float* __restrict__ msum, const float* __restrict__ deg,
    const float* __restrict__ xin, const float* __restrict__ wl,
    const float* __restrict__ bias, const float* __restrict__ wr,
    float* __restrict__ out, int nNodes) {
  __shared__ float s_wl[D * D];
  __shared__ float s_wr[D * D];
  for (int i = threadIdx.x; i < D * D; i += blockDim.x) {
    s_wl[i] = wl[i];
    s_wr[i] = wr[i];
  }
  __syncthreads();

  const int lane = threadIdx.x & 31;
  const int wave = threadIdx.x >> 5;
  const int tile = blockIdx.x * WAVES_PER_BLOCK + wave;
  if (tile * 16 >= nNodes) return;   // wave-uniform: EXEC stays all-1s

  const int row0 = tile * 16;
  const int half = lane >> 4;        // which lane half (K-split)
  const int l16 = lane & 15;
  int arow = row0 + l16;             // this lane's A-matrix row
  if (arow >= nNodes) arow = nNodes - 1;  // clamp loads for ragged tail

  const float inv = 1.0f / fmaxf(deg[arow], 1.0f);
  const float* mrow = msum + (size_t)arow * D;
  const float* xrow = xin + (size_t)arow * D;

  v8f acc[4] = {v8f{}, v8f{}, v8f{}, v8f{}};

#pragma unroll
  for (int kt = 0; kt < D / 4; ++kt) {
    const int kb = kt * 4 + half * 2;
    v2f aM, aX;
    aM.x = mrow[kb] * inv;           // fuse the mean (divide by degree)
    aM.y = mrow[kb + 1] * inv;
    aX.x = xrow[kb];
    aX.y = xrow[kb + 1];
#pragma unroll
    for (int nt = 0; nt < 4; ++nt) {
      const int col = nt * 16 + l16;
      v2f bL, bR;
      bL.x = s_wl[kb * D + col];
      bL.y = s_wl[(kb + 1) * D + col];
      bR.x = s_wr[kb * D + col];
      bR.y = s_wr[(kb + 1) * D + col];
      acc[nt] = __builtin_amdgcn_wmma_f32_16x16x4_f32(
          false, aM, false, bL, (short)0, acc[nt], false, false);
      acc[nt] = __builtin_amdgcn_wmma_f32_16x16x4_f32(
          false, aX, false, bR, (short)0, acc[nt], false, false);
    }
  }

#pragma unroll
  for (int nt = 0; nt < 4; ++nt) {
    const int col = nt * 16 + l16;
    const float b = bias[col];
#pragma unroll
    for (int r = 0; r < 8; ++r) {
      const int row = row0 + half * 8 + r;
      float v = acc[nt][r] + b;
      if (RELU) v = fmaxf(v, 0.0f);
      if (row < nNodes) out[(size_t)row * D + col] = v;
    }
  }
}

// ---------------------------------------------------------------------------
// Inputs (setup_inputs order): x, edge_index(int64, [2,E]), w1_l, b1, w1_r,
//                              w2_l, b2, w2_r. Output: [N,64] fp32.
// Workspace layout: msum [N*64] f32, deg [N] f32 (~26MB).
// Layer-1 activations h live in d_out (safe: in the layer-2 GEMM each 16-row
// tile of h is read only by the wave that later writes those same rows).
// ---------------------------------------------------------------------------
extern "C" void kernel_launch(void* const* d_in, const int* in_sizes, int n_in,
                              void* d_out, int out_size, void* d_ws, size_t ws_size,
                              hipStream_t stream) {
  const float* x = (const float*)d_in[0];
  const long long* ei = (const long long*)d_in[1];
  const float* w1l = (const float*)d_in[2];
  const float* b1 = (const float*)d_in[3];
  const float* w1r = (const float*)d_in[4];
  const float* w2l = (const float*)d_in[5];
  const float* b2 = (const float*)d_in[6];
  const float* w2r = (const float*)d_in[7];
  float* out = (float*)d_out;

  const long long nNodes = (long long)in_sizes[0] / D;
  const long long nEdges = (long long)in_sizes[1] / 2;
  const long long* src = ei;
  const long long* dst = ei + nEdges;

  float* msum = (float*)d_ws;
  float* deg = msum + nNodes * D;
  float* h = out;  // layer-1 activations

  const long long zn = nNodes * (D + 1);
  dim3 zgrid((unsigned)((zn + 255) / 256));
  dim3 sgrid((unsigned)((nEdges * D + 255) / 256));
  const long long tiles = (nNodes + 15) / 16;
  dim3 ggrid((unsigned)((tiles + WAVES_PER_BLOCK - 1) / WAVES_PER_BLOCK));

  // ---- Layer 1: h = relu(mean(x) @ w1_l + b1 + x @ w1_r) ----
  zero_f32<<<zgrid, 256, 0, stream>>>(msum, zn);
  scatter_mean_accum<<<sgrid, 256, 0, stream>>>(x, src, dst, msum, deg, nEdges);
  sage_wmma_gemm<true><<<ggrid, 256, 0, stream>>>(msum, deg, x, w1l, b1, w1r, h,
                                                  (int)nNodes);

  // ---- Layer 2: out = mean(h) @ w2_l + b2 + h @ w2_r ----
  zero_f32<<<zgrid, 256, 0, stream>>>(msum, zn);
  scatter_mean_accum<<<sgrid, 256, 0, stream>>>(h, src, dst, msum, deg, nEdges);
  sage_wmma_gemm<false><<<ggrid, 256, 0, stream>>>(msum, deg, h, w2l, b2, w2r,
                                                   out, (int)nNodes);
}